// MaskDiffusion_74311524155684
// MI455X (gfx1250) — compile-verified
//
#include <hip/hip_runtime.h>
#include <math.h>

// Problem constants (match reference)
#define MASK_ID   103
#define TIMESTEPS 200
#define BB        16
#define TT        512
#define VV        32000

#define WAVES_PER_BLOCK 8
#define DEPTH           4       // async pipeline depth (power of 2)
#define STAGES          250     // 32000 floats / (32 lanes * 4 floats per lane)

// ---------------- CDNA5 async global->LDS helpers (inline asm) ----------------
__device__ __forceinline__ void async_load_b128(unsigned lds_addr, const void* gaddr) {
  // GLOBAL_LOAD_ASYNC_TO_LDS_B128: VDST = VGPR holding LDS byte address,
  // VADDR = 64-bit global address VGPR pair, no SADDR ("off"). Tracked by ASYNCcnt.
  asm volatile("global_load_async_to_lds_b128 %0, %1, off"
               :: "v"(lds_addr), "v"(gaddr)
               : "memory");
}

template <int N>
__device__ __forceinline__ void wait_asynccnt() {
  asm volatile("s_wait_asynccnt %0" :: "n"(N) : "memory");
}

// ---------------- Kernel 1: per-row online softmax-max/argmax/sum -------------
// One wave32 per row of 32000 logits. 4-deep async double-buffer of
// 512B tiles (32 lanes x float4) streamed through LDS.
__global__ __launch_bounds__(256) void rowstat_kernel(const float* __restrict__ logits,
                                                      float* __restrict__ conf_out,
                                                      int* __restrict__ pred_out) {
  __shared__ float4 tiles[WAVES_PER_BLOCK][DEPTH][32];

  const int wave = threadIdx.x >> 5;
  const int lane = threadIdx.x & 31;
  const int row  = blockIdx.x * WAVES_PER_BLOCK + wave;   // 0..8191

  const float* gp = logits + (size_t)row * VV + lane * 4;

  unsigned lds_base[DEPTH];
#pragma unroll
  for (int d = 0; d < DEPTH; ++d) {
    // low 32 bits of the generic pointer to a __shared__ object == LDS byte addr
    lds_base[d] = (unsigned)(size_t)(&tiles[wave][d][lane]);
  }

  // Prologue: fill pipeline (stages 0..2)
  async_load_b128(lds_base[0], gp);
  async_load_b128(lds_base[1], gp + 128);
  async_load_b128(lds_base[2], gp + 2 * 128);

  const float L2E = 1.4426950408889634f;
  float m = -INFINITY;   // running row max
  float s = 0.0f;        // running sum exp(l - m)
  int   idx = 0;         // running argmax (first occurrence)

  auto consume = [&](int k) {
    float4 v = tiles[wave][k & (DEPTH - 1)][lane];
    const int base = k * 128 + lane * 4;
    const float m0 = m;
    // strict > keeps the first occurrence within this lane's increasing indices
    if (v.x > m) { m = v.x; idx = base;     }
    if (v.y > m) { m = v.y; idx = base + 1; }
    if (v.z > m) { m = v.z; idx = base + 2; }
    if (v.w > m) { m = v.w; idx = base + 3; }
    s = s * exp2f((m0 - m) * L2E)
      + exp2f((v.x - m) * L2E) + exp2f((v.y - m) * L2E)
      + exp2f((v.z - m) * L2E) + exp2f((v.w - m) * L2E);
  };

  // Steady state: issue k+3, wait so stage k has landed (<=3 outstanding), consume k.
  for (int k = 0; k < STAGES - 3; ++k) {
    async_load_b128(lds_base[(k + 3) & (DEPTH - 1)], gp + (size_t)(k + 3) * 128);
    wait_asynccnt<3>();
    consume(k);
  }
  wait_asynccnt<2>(); consume(STAGES - 3);
  wait_asynccnt<1>(); consume(STAGES - 2);
  wait_asynccnt<0>(); consume(STAGES - 1);

  // Wave32 butterfly reduction; min-index tie-break to match jnp.argmax.
#pragma unroll
  for (int off = 16; off >= 1; off >>= 1) {
    float om = __shfl_xor(m, off, 32);
    float os = __shfl_xor(s, off, 32);
    int   oi = __shfl_xor(idx, off, 32);
    float M  = fmaxf(m, om);
    s = s * exp2f((m - M) * L2E) + os * exp2f((om - M) * L2E);
    if (om > m || (om == m && oi < idx)) idx = oi;
    m = M;
  }

  if (lane == 0) {
    conf_out[row] = 1.0f / s;   // max softmax prob = exp(0)/sum
    pred_out[row] = idx;
  }
}

// ---------------- Kernel 2: rank-based re-masking per batch row ---------------
// One block of 512 threads per batch row. Stable selection of the
// n_mask_prev lexicographically-smallest (conf, index) eligible positions.
__global__ __launch_bounds__(512) void remask_kernel(const int* __restrict__ target_ids,
                                                     const int* __restrict__ gen_mask,
                                                     const int* __restrict__ t_ptr,
                                                     const float* __restrict__ conf,
                                                     const int* __restrict__ pred,
                                                     float* __restrict__ out_tokens) {
  __shared__ float cs[TT];
  __shared__ int   s_ngen;

  const int b = blockIdx.x;
  const int i = threadIdx.x;
  const int g = b * TT + i;

  if (i == 0) s_ngen = 0;
  __syncthreads();

  const int   id     = target_ids[g];
  const int   gm     = gen_mask[g];
  const bool  masked = (id == MASK_ID);
  const float c      = conf[g];

  cs[i] = (gm != 0 && masked) ? c : INFINITY;   // ineligible -> +inf (sorts last, stable)
  if (gm != 0) atomicAdd(&s_ngen, 1);
  __syncthreads();

  // gamma_prev = clip(1 - cos(pi/2 * (t-1)/TIMESTEPS)^2, 0, 1)
  const int   t  = t_ptr[0];
  const float a  = (1.5707963267948966f * (float)(t - 1)) / (float)TIMESTEPS;
  const float cp = cosf(a);
  float gamma_prev = 1.0f - cp * cp;
  gamma_prev = fminf(fmaxf(gamma_prev, 0.0f), 1.0f);

  int n_mask_prev = (int)(gamma_prev * (float)s_ngen);  // astype(int32) truncates
  if (n_mask_prev < 0) n_mask_prev = 0;                 // jnp.maximum(,0)

  // Stable rank of (cs[i], i): count strictly-smaller keys.
  const float ci = cs[i];
  int rank = 0;
  for (int q = 0; q < TT; ++q) {
    const float cq = cs[q];
    rank += (int)((cq < ci) || (cq == ci && q < i));
  }
  const bool remask = rank < n_mask_prev;

  int newid = masked ? pred[g] : id;   // unmask everything masked
  if (remask) newid = MASK_ID;         // re-mask least confident

  out_tokens[g] = (float)newid;
}

// --------------------------------- launcher ----------------------------------
extern "C" void kernel_launch(void* const* d_in, const int* in_sizes, int n_in,
                              void* d_out, int out_size, void* d_ws, size_t ws_size,
                              hipStream_t stream) {
  (void)in_sizes; (void)n_in; (void)out_size; (void)ws_size;

  const float* logits     = (const float*)d_in[0];  // (B,T,V) f32
  const int*   target_ids = (const int*)d_in[1];    // (B,T) i32
  const int*   gen_mask   = (const int*)d_in[2];    // (B,T) bool->i32
  const int*   t_ptr      = (const int*)d_in[3];    // scalar i32

  float* out_tokens = (float*)d_out;                // output[0]: new_target (B,T)
  float* out_conf   = (float*)d_out + BB * TT;      // output[1]: confidence (B,T)
  int*   pred       = (int*)d_ws;                   // scratch: argmax tokens (B,T)

  // 8192 rows, one wave32 per row, 8 waves per block.
  rowstat_kernel<<<(BB * TT) / WAVES_PER_BLOCK, 32 * WAVES_PER_BLOCK, 0, stream>>>(
      logits, out_conf, pred);

  remask_kernel<<<BB, TT, 0, stream>>>(target_ids, gen_mask, t_ptr, out_conf, pred,
                                       out_tokens);
}